// SAGPool_61211873902597
// MI455X (gfx1250) — compile-verified
//
#include <hip/hip_runtime.h>
#include <hip/hip_bf16.h>

typedef __attribute__((ext_vector_type(2))) float v2f;
typedef __attribute__((ext_vector_type(8))) float v8f;

#define C_DIM 256
#define NPG   256
#define TOPK  205
#define LDS_STRIDE 272  // 16B-aligned, rows k and k+2 land on disjoint bank groups

__global__ void zero_f32(float* p, int n) {
  int i = blockIdx.x * blockDim.x + threadIdx.x;
  if (i < n) p[i] = 0.0f;
}

// h[i] = dot(x[i,:], W[:,0]); one wave32 per node, coalesced 128B strides.
__global__ void __launch_bounds__(256) gemv_h(const float* __restrict__ x,
                                              const float* __restrict__ W,
                                              float* __restrict__ h, int n) {
  __shared__ float w_s[C_DIM];
  int t = threadIdx.x;
  w_s[t] = W[t];
  __syncthreads();
  int lane = t & 31;
  int wave = t >> 5;
  int node = blockIdx.x * 8 + wave;
  if (node >= n) return;
  const float* xr = x + (size_t)node * C_DIM;
  float s = 0.0f;
#pragma unroll
  for (int it = 0; it < C_DIM / 32; ++it) {
    int c = lane + it * 32;
    s += xr[c] * w_s[c];
  }
#pragma unroll
  for (int off = 16; off > 0; off >>= 1) s += __shfl_down(s, off, 32);
  if (lane == 0) h[node] = s;
}

// deg[col] += 1 (exact: small-integer float adds are order-invariant)
__global__ void deg_count(const int* __restrict__ col, float* __restrict__ deg, int E) {
  int e = blockIdx.x * blockDim.x + threadIdx.x;
  if (e < E) atomicAdd(&deg[col[e]], 1.0f);
}

// in-place deg -> rsqrt(deg + 1)  (self-loop)
__global__ void make_dinv(float* dd, int n) {
  int i = blockIdx.x * blockDim.x + threadIdx.x;
  if (i < n) dd[i] = rsqrtf(dd[i] + 1.0f);
}

// acc[col] += dinv[row] * h[row]
__global__ void edge_scatter(const int* __restrict__ row, const int* __restrict__ col,
                             const float* __restrict__ dinv, const float* __restrict__ h,
                             float* __restrict__ acc, int E) {
  int e = blockIdx.x * blockDim.x + threadIdx.x;
  if (e < E) {
    int r = row[e];
    atomicAdd(&acc[col[e]], dinv[r] * h[r]);
  }
}

// score[i] = dinv[i]*(acc[i] + dinv[i]*h[i]) + b   (in place into acc)
__global__ void finalize_score(const float* __restrict__ dinv, const float* __restrict__ h,
                               float* __restrict__ acc, const float* __restrict__ b, int n) {
  int i = blockIdx.x * blockDim.x + threadIdx.x;
  if (i < n) {
    float di = dinv[i];
    acc[i] = di * (acc[i] + di * h[i]) + b[0];
  }
}

// Per-graph top-K via stable rank: rank_i = #{j: s_j > s_i} + #{j<i: s_j == s_i}.
// Ranks form a permutation of 0..255 -> exactly K kept, matches lax.top_k semantics.
__global__ void __launch_bounds__(NPG) topk_gate(const float* __restrict__ score,
                                                 float* __restrict__ gate,
                                                 int* __restrict__ sel) {
  __shared__ float s_s[NPG];
  int i = threadIdx.x;
  int node = blockIdx.x * NPG + i;
  float si = score[node];
  s_s[i] = si;
  __syncthreads();
  int r = 0;
  for (int j = 0; j < NPG; ++j) {
    float sj = s_s[j];
    r += (sj > si) || (sj == si && j < i) ? 1 : 0;
  }
  bool keep = r < TOPK;
  gate[node] = keep ? tanhf(si) : 0.0f;  // zero-gate drops node from the mean
  sel[node]  = keep ? 1 : 0;             // mask drops node from the max
}

// One block (8 waves) per graph. Wave w owns columns [32w, 32w+32) = two N=16 tiles.
// Mean pool via V_WMMA_F32_16X16X4_F32: A rows all = gates of the k-quad, B = LDS-staged
// x tile. Max pool reuses the exact B-fragment scalars (zero extra traffic).
__global__ void __launch_bounds__(256) pool_wmma(const float* __restrict__ x,
                                                 const float* __restrict__ gate,
                                                 const int* __restrict__ sel,
                                                 float* __restrict__ out) {
  __shared__ float tile[16 * LDS_STRIDE];
  __shared__ float g_s[NPG];
  __shared__ int   sel_s[NPG];
  int gidx = blockIdx.x;
  int t = threadIdx.x;
  int lane = t & 31;
  int wave = t >> 5;
  size_t base = (size_t)gidx * NPG;
  g_s[t] = gate[base + t];
  sel_s[t] = sel[base + t];

  int n0a = wave * 32;
  int n0b = wave * 32 + 16;
  int lcol = lane & 15;
  int hoff = (lane < 16) ? 0 : 2;  // K-row offset for this half-wave

  v8f cA = {};
  v8f cB = {};
  float mxA = -__builtin_inff();
  float mxB = -__builtin_inff();
  const float* xg = x + base * C_DIM;

  for (int chunk = 0; chunk < NPG / 16; ++chunk) {
    __syncthreads();
    const float* src = xg + (size_t)chunk * 16 * C_DIM;
#pragma unroll
    for (int r = 0; r < 16; ++r)  // 256 threads: 1KB coalesced per row
      tile[r * LDS_STRIDE + t] = src[r * C_DIM + t];
    __syncthreads();

#pragma unroll
    for (int kq = 0; kq < 4; ++kq) {
      int k = kq * 4;
      int krow = chunk * 16 + k;
      // A fragment: gates, identical across M rows (broadcast LDS reads)
      v2f a;
      a.x = g_s[krow + hoff];
      a.y = g_s[krow + hoff + 1];
      // B fragments: VGPR0 holds K rows {0,2}, VGPR1 holds {1,3} across half-waves
      const float* r0 = &tile[(k + hoff) * LDS_STRIDE];
      const float* r1 = &tile[(k + hoff + 1) * LDS_STRIDE];
      v2f bA, bB;
      bA.x = r0[n0a + lcol];  bA.y = r1[n0a + lcol];
      bB.x = r0[n0b + lcol];  bB.y = r1[n0b + lcol];
      // max pool over selected rows, reusing B scalars
      int s0 = sel_s[krow + hoff];
      int s1 = sel_s[krow + hoff + 1];
      if (s0) { mxA = fmaxf(mxA, a.x * bA.x); mxB = fmaxf(mxB, a.x * bB.x); }
      if (s1) { mxA = fmaxf(mxA, a.y * bA.y); mxB = fmaxf(mxB, a.y * bB.y); }
      cA = __builtin_amdgcn_wmma_f32_16x16x4_f32(false, a, false, bA, (short)0, cA, false, false);
      cB = __builtin_amdgcn_wmma_f32_16x16x4_f32(false, a, false, bB, (short)0, cB, false, false);
    }
  }
  // node rows are split across half-waves for the max; fold them
  mxA = fmaxf(mxA, __shfl_xor(mxA, 16, 32));
  mxB = fmaxf(mxB, __shfl_xor(mxB, 16, 32));

  if (lane < 16) {  // D row M=0 lives in c[0] of lanes 0..15
    const float inv_k = 1.0f / (float)TOPK;
    float* og = out + (size_t)gidx * (2 * C_DIM);
    og[n0a + lane] = cA[0] * inv_k;
    og[n0b + lane] = cB[0] * inv_k;
    og[C_DIM + n0a + lane] = mxA;
    og[C_DIM + n0b + lane] = mxB;
  }
}

extern "C" void kernel_launch(void* const* d_in, const int* in_sizes, int n_in,
                              void* d_out, int out_size, void* d_ws, size_t ws_size,
                              hipStream_t stream) {
  const float* x = (const float*)d_in[0];
  const int*   ei = (const int*)d_in[1];   // [2, E] row-major
  const float* W = (const float*)d_in[4];
  const float* b = (const float*)d_in[5];
  float* out = (float*)d_out;

  int n = in_sizes[0] / C_DIM;   // 131072
  int E = in_sizes[1] / 2;       // 2097152
  int G = n / NPG;               // 512
  const int* erow = ei;
  const int* ecol = ei + E;

  float* h    = (float*)d_ws;    // [n]
  float* dinv = h + n;           // [n] deg -> dinv in place
  float* acc  = dinv + n;        // [n] acc -> score in place
  float* gate = acc + n;         // [n]
  int*   sel  = (int*)(gate + n);// [n]

  zero_f32<<<(2 * n + 255) / 256, 256, 0, stream>>>(dinv, 2 * n);  // deg + acc contiguous
  gemv_h<<<(n + 7) / 8, 256, 0, stream>>>(x, W, h, n);
  deg_count<<<(E + 255) / 256, 256, 0, stream>>>(ecol, dinv, E);
  make_dinv<<<(n + 255) / 256, 256, 0, stream>>>(dinv, n);
  edge_scatter<<<(E + 255) / 256, 256, 0, stream>>>(erow, ecol, dinv, h, acc, E);
  finalize_score<<<(n + 255) / 256, 256, 0, stream>>>(dinv, h, acc, b, n);
  topk_gate<<<G, NPG, 0, stream>>>(acc, gate, sel);
  pool_wmma<<<G, 256, 0, stream>>>(x, gate, sel, out);
}